// InvariantMessagePasser_18468359373228
// MI455X (gfx1250) — compile-verified
//
#include <hip/hip_runtime.h>
#include <hip/hip_bf16.h>

#define NSH 16          // (L_MAX+1)^2
#define K_DIM 32
#define RB_ROW 128      // (L_MAX+1)*K
#define MP_SCALING 0.1f

typedef __attribute__((ext_vector_type(2))) float v2f;
typedef __attribute__((ext_vector_type(8))) float v8f;

__device__ __forceinline__ int l_of_m(int m) {
    return (m == 0) ? 0 : (m < 4) ? 1 : (m < 9) ? 2 : 3;
}

// ---------------- phase 0: zero int scratch ----------------
__global__ void zero_i32(int* __restrict__ p, int n) {
    int i = blockIdx.x * blockDim.x + threadIdx.x;
    if (i < n) p[i] = 0;
}

// ---------------- phase 1: per-atom edge histogram ----------------
__global__ void hist_kernel(const int* __restrict__ centers, int* __restrict__ counts, int E) {
    int e = blockIdx.x * blockDim.x + threadIdx.x;
    if (e < E) atomicAdd(&counts[centers[e]], 1);
}

// ---------------- phase 2: exclusive scan (single block, chunked) ----------------
__global__ void scan_kernel(const int* __restrict__ counts, int* __restrict__ offsets, int n) {
    __shared__ int sdata[1024];
    __shared__ int s_running;
    int tid = threadIdx.x;
    if (tid == 0) s_running = 0;
    __syncthreads();
    for (int base = 0; base < n; base += 1024) {
        int v = (base + tid < n) ? counts[base + tid] : 0;
        sdata[tid] = v;
        __syncthreads();
        for (int off = 1; off < 1024; off <<= 1) {
            int t = 0;
            if (tid >= off) t = sdata[tid - off];
            __syncthreads();
            if (tid >= off) sdata[tid] += t;
            __syncthreads();
        }
        int incl = sdata[tid];
        int r = s_running;
        __syncthreads();
        if (base + tid < n) offsets[base + tid] = r + incl - v;  // exclusive
        if (tid == 0) s_running = r + sdata[1023];
        __syncthreads();
    }
    if (tid == 0) offsets[n] = s_running;
}

// ---------------- phase 3: fill CSR edge list ----------------
__global__ void fill_kernel(const int* __restrict__ centers,
                            const int* __restrict__ offsets,
                            int* __restrict__ cursor,
                            int* __restrict__ edge_list, int E) {
    int e = blockIdx.x * blockDim.x + threadIdx.x;
    if (e < E) {
        int c = centers[e];
        int pos = atomicAdd(&cursor[c], 1);
        edge_list[offsets[c] + pos] = e;
    }
}

// ---------------- phase 4: one wave32 per atom, WMMA f32 16x16x4 accumulation ----------------
// Per edge:  D(16x16) += A(16x4) x B(4x16), K-dim = 4 l-channels.
//   A[m,l] = sh[e,m] * (l(m)==l)          (lane-static mask)
//   B[l,k] = rb[e,l,k] * emb[nb,k]        (two k-halves -> two WMMAs)
// A layout (ISA 7.12.2): lanes0-15 M=lane, v.x=K0 v.y=K1; lanes16-31 v.x=K2 v.y=K3.
// B layout mirrors (rows K in VGPRs, cols N across lanes, lanes16-31 hold K=2,3).
// D layout: VGPR r -> row r (lanes0-15) / row r+8 (lanes16-31), col = lane&15.
__global__ void __launch_bounds__(256)
atom_wmma_kernel(const float* __restrict__ sh,
                 const float* __restrict__ rb,
                 const float* __restrict__ emb,
                 const int* __restrict__ neighbors,
                 const int* __restrict__ offsets,
                 const int* __restrict__ edge_list,
                 float* __restrict__ out, int n_atoms) {
    int wave = (int)((blockIdx.x * blockDim.x + threadIdx.x) >> 5);
    int lane = threadIdx.x & 31;
    if (wave >= n_atoms) return;               // wave-uniform: EXEC stays all-ones
    int a = wave;
    int beg = offsets[a];
    int end = offsets[a + 1];

    int half = lane >> 4;        // lane group: 0 -> l{0,1}, 1 -> l{2,3}
    int lb   = half * 2;         // l base for this lane group
    int m    = lane & 15;        // SH row handled by this lane (A matrix)
    int kq   = lane & 15;        // k column within a 16-wide half (B/D matrices)
    int lm   = l_of_m(m);
    float selx = (lm == lb)     ? 1.0f : 0.0f;
    float sely = (lm == lb + 1) ? 1.0f : 0.0f;

    v8f d0 = {};  // k = 0..15
    v8f d1 = {};  // k = 16..31

    for (int i = beg; i < end; ++i) {
        int e  = edge_list[i];
        int nb = neighbors[e];
        const float* rbe  = rb  + (long)e  * RB_ROW;
        const float* embn = emb + (long)nb * K_DIM;

        float s = sh[(long)e * NSH + m];
        v2f A;
        A.x = selx * s;
        A.y = sely * s;

        {   // k-half 0
            float ev = embn[kq];
            v2f B;
            B.x = rbe[lb * K_DIM + kq] * ev;
            B.y = rbe[(lb + 1) * K_DIM + kq] * ev;
            d0 = __builtin_amdgcn_wmma_f32_16x16x4_f32(false, A, false, B,
                                                       (short)0, d0, false, false);
        }
        {   // k-half 1
            float ev = embn[kq + 16];
            v2f B;
            B.x = rbe[lb * K_DIM + kq + 16] * ev;
            B.y = rbe[(lb + 1) * K_DIM + kq + 16] * ev;
            d1 = __builtin_amdgcn_wmma_f32_16x16x4_f32(false, A, false, B,
                                                       (short)0, d1, false, false);
        }
    }

    float* outa = out + (long)a * (NSH * K_DIM);
    int mbase = half * 8;
#pragma unroll
    for (int r = 0; r < 8; ++r) {
        outa[(mbase + r) * K_DIM + kq]      = d0[r] * MP_SCALING;
        outa[(mbase + r) * K_DIM + kq + 16] = d1[r] * MP_SCALING;
    }
}

// ---------------- fallback: atomic scatter (if ws too small) ----------------
__global__ void zero_f32(float* __restrict__ p, long n) {
    long i = blockIdx.x * (long)blockDim.x + threadIdx.x;
    if (i < n) p[i] = 0.0f;
}

__global__ void scatter_atomic_kernel(const float* __restrict__ sh,
                                      const float* __restrict__ rb,
                                      const float* __restrict__ emb,
                                      const int* __restrict__ centers,
                                      const int* __restrict__ neighbors,
                                      float* __restrict__ out, int E) {
    long t = blockIdx.x * (long)blockDim.x + threadIdx.x;
    int e = (int)(t >> 5);
    int k = (int)(t & 31);
    if (e >= E) return;
    int c = centers[e], nb = neighbors[e];
    float ev = emb[(long)nb * K_DIM + k];
    const float* rbe = rb + (long)e * RB_ROW;
    const float* she = sh + (long)e * NSH;
    float w[4];
#pragma unroll
    for (int l = 0; l < 4; ++l) w[l] = rbe[l * K_DIM + k] * ev;
    float* o = out + (long)c * (NSH * K_DIM) + k;
#pragma unroll
    for (int m = 0; m < NSH; ++m) {
        atomicAdd(o + m * K_DIM, MP_SCALING * she[m] * w[l_of_m(m)]);
    }
}

extern "C" void kernel_launch(void* const* d_in, const int* in_sizes, int n_in,
                              void* d_out, int out_size, void* d_ws, size_t ws_size,
                              hipStream_t stream) {
    const float* sh        = (const float*)d_in[0];
    const float* rb        = (const float*)d_in[1];
    const float* emb       = (const float*)d_in[2];
    const int*   centers   = (const int*)d_in[3];
    const int*   neighbors = (const int*)d_in[4];

    int E       = in_sizes[0] / NSH;
    int n_atoms = out_size / (NSH * K_DIM);

    size_t need = ((size_t)3 * n_atoms + 1 + (size_t)E) * sizeof(int);
    if (ws_size >= need) {
        int* counts    = (int*)d_ws;
        int* cursor    = counts + n_atoms;
        int* offsets   = cursor + n_atoms;          // n_atoms + 1 entries
        int* edge_list = offsets + n_atoms + 1;     // E entries

        zero_i32<<<(2 * n_atoms + 255) / 256, 256, 0, stream>>>(counts, 2 * n_atoms);
        hist_kernel<<<(E + 255) / 256, 256, 0, stream>>>(centers, counts, E);
        scan_kernel<<<1, 1024, 0, stream>>>(counts, offsets, n_atoms);
        fill_kernel<<<(E + 255) / 256, 256, 0, stream>>>(centers, offsets, cursor, edge_list, E);

        const int waves_per_block = 8;  // 256 threads = 8 wave32s
        int blocks = (n_atoms + waves_per_block - 1) / waves_per_block;
        atom_wmma_kernel<<<blocks, 256, 0, stream>>>(sh, rb, emb, neighbors,
                                                     offsets, edge_list,
                                                     (float*)d_out, n_atoms);
    } else {
        long n = (long)out_size;
        zero_f32<<<(int)((n + 255) / 256), 256, 0, stream>>>((float*)d_out, n);
        long tot = (long)E * 32;
        scatter_atomic_kernel<<<(int)((tot + 255) / 256), 256, 0, stream>>>(
            sh, rb, emb, centers, neighbors, (float*)d_out, E);
    }
}